// PointerNetworkRL_29231547417196
// MI455X (gfx1250) — compile-verified
//
#include <hip/hip_runtime.h>
#include <math.h>

#define NNODES 100
#define NPAD   112
#define HDIM   64
#define BATCH  256
#define STEPS  200
#define NEGV   (-1.0e9f)
#define PB     (NPAD * HDIM)   // 7168 floats per batch matrix

typedef __attribute__((ext_vector_type(2))) float v2f;
typedef __attribute__((ext_vector_type(8))) float v8f;

// D[16*Mt x 16*Nt] = scale * (A[Mx K] x B) + bias[n]
// B element (k,n) = B[k*ldbK + n*ldbN]  (lets us do K^T with ldbK=1, ldbN=ld)
// One wave per 16x16 tile; 8 waves per block. V_WMMA_F32_16X16X4_F32, K step 4.
__device__ __forceinline__ void mm_wmma_f32(
    const float* A, int lda,
    const float* B, int ldbK, int ldbN,
    float* D, int ldd,
    const float* bias, float scale,
    int Mt, int Nt, int Kdim,
    int wave, int lane)
{
    const int mr = lane & 15;              // M row (A) / N col (B,D) within tile
    const int kh = (lane >> 4) << 1;       // lanes 16-31 hold K=2,3
    const int ntiles = Mt * Nt;
    for (int t = wave; t < ntiles; t += 8) {
        const int m0 = (t / Nt) * 16;
        const int n0 = (t % Nt) * 16;
        v8f acc = {};
        for (int k0 = 0; k0 < Kdim; k0 += 4) {
            v2f a, bb;
            const float* ap = A + (m0 + mr) * lda + (k0 + kh);
            a.x = ap[0];
            a.y = ap[1];
            const float* bp = B + (size_t)(k0 + kh) * (size_t)ldbK
                                + (size_t)(n0 + mr) * (size_t)ldbN;
            bb.x = bp[0];
            bb.y = bp[ldbK];
            acc = __builtin_amdgcn_wmma_f32_16x16x4_f32(
                false, a, false, bb, (short)0, acc, false, false);
        }
        const int n  = n0 + mr;
        const int mb = m0 + ((lane >> 4) << 3);   // lanes 16-31 hold rows M=8..15
        const float bv = bias ? bias[n] : 0.0f;
#pragma unroll
        for (int i = 0; i < 8; ++i)
            D[(mb + i) * ldd + n] = acc[i] * scale + bv;
    }
}

// ---------------- Encoder: emb -> Q,K,V -> softmax(QK^T/8) -> attended -> A1
__global__ __launch_bounds__(256) void ptrnet_encoder(
    const float* __restrict__ coords, const float* __restrict__ demands,
    const float* __restrict__ We_w, const float* __restrict__ We_b,
    const float* __restrict__ Wq_w, const float* __restrict__ Wq_b,
    const float* __restrict__ Wk_w, const float* __restrict__ Wk_b,
    const float* __restrict__ Wv_w, const float* __restrict__ Wv_b,
    const float* __restrict__ P1_w, const float* __restrict__ P1_b,
    float* __restrict__ gQ, float* __restrict__ gK, float* __restrict__ gV,
    float* __restrict__ gAtt, float* __restrict__ gA1)
{
    __shared__ float s_pool[NPAD * NPAD];   // 12544 floats = 50KB; emb then S
    const int b    = blockIdx.x;
    const int tid  = threadIdx.x;
    const int lane = tid & 31;
    const int wave = tid >> 5;

    // Phase 1: emb[n][h] = nf @ We + b  (padded rows zeroed)
    for (int i = tid; i < NPAD * HDIM; i += 256) {
        const int n = i >> 6, h = i & 63;
        float v = 0.0f;
        if (n < NNODES) {
            const float x = coords[(b * NNODES + n) * 2 + 0];
            const float y = coords[(b * NNODES + n) * 2 + 1];
            const float d = demands[b * NNODES + n];
            v = fmaf(x, We_w[h], fmaf(y, We_w[64 + h], fmaf(d, We_w[128 + h], We_b[h])));
        }
        s_pool[i] = v;
    }
    __syncthreads();

    float* Qb = gQ   + (size_t)b * PB;
    float* Kb = gK   + (size_t)b * PB;
    float* Vb = gV   + (size_t)b * PB;
    float* Ab = gAtt + (size_t)b * PB;
    float* A1 = gA1  + (size_t)b * PB;

    // Phase 2: Q,K,V = emb @ W + b   (7x4 tiles, K=64)
    mm_wmma_f32(s_pool, 64, Wq_w, 64, 1, Qb, 64, Wq_b, 1.0f, 7, 4, 64, wave, lane);
    mm_wmma_f32(s_pool, 64, Wk_w, 64, 1, Kb, 64, Wk_b, 1.0f, 7, 4, 64, wave, lane);
    mm_wmma_f32(s_pool, 64, Wv_w, 64, 1, Vb, 64, Wv_b, 1.0f, 7, 4, 64, wave, lane);
    __threadfence();
    __syncthreads();

    // Phase 3: S = Q @ K^T / sqrt(64)  into LDS (emb dead), 7x7 tiles
    mm_wmma_f32(Qb, 64, Kb, /*ldbK=*/1, /*ldbN=*/64, s_pool, NPAD,
                nullptr, 0.125f, 7, 7, 64, wave, lane);
    __syncthreads();

    // Phase 4: row softmax over the 100 valid columns
    if (tid < NNODES) {
        float* row = s_pool + tid * NPAD;
        float mx = row[0];
        for (int n = 1; n < NNODES; ++n) mx = fmaxf(mx, row[n]);
        float sm = 0.0f;
        for (int n = 0; n < NNODES; ++n) { float e = expf(row[n] - mx); row[n] = e; sm += e; }
        const float inv = 1.0f / sm;
        for (int n = 0; n < NNODES; ++n) row[n] *= inv;
    }
    __syncthreads();

    // Phase 5: attended = attn @ V   (K=100, exact 25 wmma k-steps)
    mm_wmma_f32(s_pool, NPAD, Vb, 64, 1, Ab, 64, nullptr, 1.0f, 7, 4, NNODES, wave, lane);
    __threadfence();
    __syncthreads();

    // Phase 6: A1 = attended @ P1_w[0:64] + P1_b  (step-invariant decoder term)
    mm_wmma_f32(Ab, 64, P1_w, 64, 1, A1, 64, P1_b, 1.0f, 7, 4, 64, wave, lane);
}

// ---------------- Decoder: 200 sequential greedy steps, all state in LDS
#define LD65 65   // padded LDS stride: bank = (row + h) mod 64 -> conflict-free

__global__ __launch_bounds__(256) void ptrnet_decoder(
    const float* __restrict__ demands, const float* __restrict__ capacities,
    const float* __restrict__ P1_w, const float* __restrict__ P2_w,
    const float* __restrict__ gAtt, const float* __restrict__ gA1,
    int* __restrict__ out_actions, float* __restrict__ out_logp)
{
    __shared__ float s_att[NPAD * LD65];
    __shared__ float s_A1 [NPAD * LD65];
    __shared__ float s_ctx[HDIM], s_C1[HDIM], s_P2[HDIM];
    __shared__ float s_scores[NPAD];
    __shared__ float s_dem[NPAD];
    __shared__ float s_red[256];
    __shared__ int   s_redi[128];
    __shared__ int   s_vis[NPAD];
    __shared__ int   s_cnt_all, s_cnt_cust, s_cnt_mask, s_cur, s_done;
    __shared__ float s_rem, s_logp, s_sumexp, s_raw0, s_capv;

    const int b   = blockIdx.x;
    const int tid = threadIdx.x;
    const float* P1b = P1_w + 64 * 64;   // rows 64..127 of P1_w

    // Load per-batch tensors into LDS (padded stride 65)
    for (int i = tid; i < NNODES * HDIM; i += 256) {
        const int n = i >> 6, h = i & 63;
        s_att[n * LD65 + h] = gAtt[(size_t)b * PB + n * HDIM + h];
        s_A1 [n * LD65 + h] = gA1 [(size_t)b * PB + n * HDIM + h];
    }
    if (tid < HDIM)  s_P2[tid]  = P2_w[tid];
    if (tid < NPAD)  s_vis[tid] = 0;
    if (tid < NNODES) s_dem[tid] = demands[b * NNODES + tid];
    if (tid == 0) {
        s_capv = capacities[b];
        s_rem  = s_capv;
        s_cur  = 0;
        s_done = 0;
        s_logp = 0.0f;
    }

    for (int step = 0; step < STEPS; ++step) {
        __syncthreads();                       // prev-step state visible
        const int cur0 = s_cur;
        if (tid == 0) { s_cnt_all = 0; s_cnt_cust = 0; s_cnt_mask = 0; s_sumexp = 0.0f; }
        __syncthreads();

        // unvisited counts
        if (tid < NNODES) {
            if (!s_vis[tid]) {
                atomicAdd(&s_cnt_all, 1);
                if (tid >= 1) atomicAdd(&s_cnt_cust, 1);
            }
        }
        __syncthreads();
        if (tid == 0) {
            if (s_cnt_cust == 0 && cur0 == 0) s_done = 1;   // done |= all-visited & at-depot
        }

        // ctx = mean(attended over unvisited): 4 n-groups x 64 h
        {
            const int h = tid & 63, g = tid >> 6;
            float p = 0.0f;
            for (int n = g; n < NNODES; n += 4)
                if (!s_vis[n]) p += s_att[n * LD65 + h];
            s_red[tid] = p;
        }
        __syncthreads();
        if (tid < HDIM) {
            float c = s_red[tid] + s_red[tid + 64] + s_red[tid + 128] + s_red[tid + 192];
            float cnt = (float)s_cnt_all;
            if (cnt < 1.0f) cnt = 1.0f;
            s_ctx[tid] = c / cnt;
        }
        __syncthreads();
        // C1 = ctx @ P1_w[64:128]
        if (tid < HDIM) {
            float acc = 0.0f;
            for (int k = 0; k < HDIM; ++k)
                acc = fmaf(s_ctx[k], P1b[k * HDIM + tid], acc);
            s_C1[tid] = acc;
        }
        __syncthreads();

        // scores[n] = relu(A1[n] + C1) . P2  (P2_b shift cancels in log_softmax)
        if (tid < NNODES) {
            float sc = 0.0f;
            const float* a1 = s_A1 + tid * LD65;
            for (int h = 0; h < HDIM; ++h) {
                const float pre = a1[h] + s_C1[h];
                sc = fmaf(pre > 0.0f ? pre : 0.0f, s_P2[h], sc);
            }
            int m = (s_vis[tid] != 0) | (s_dem[tid] > s_rem);
            if (tid == 0) {
                s_raw0 = sc;
                if (cur0 == 0) m = 1;        // depot masked if already at depot
            }
            if (m) atomicAdd(&s_cnt_mask, 1);
            s_scores[tid] = m ? NEGV : sc;
        }
        __syncthreads();
        if (tid == 0) {
            if (s_cnt_mask == NNODES) {      // everything masked
                if (cur0 != 0) s_scores[0] = s_raw0;   // force depot open
                else           s_done = 1;
            }
        }
        __syncthreads();

        // argmax (first-index tiebreak) over 100 scores, 128-wide tree
        if (tid < 128) {
            s_red[tid]  = (tid < NNODES) ? s_scores[tid] : -3.0e38f;
            s_redi[tid] = tid;
        }
        __syncthreads();
        for (int s = 64; s > 0; s >>= 1) {
            if (tid < s) {
                const float v2 = s_red[tid + s];
                const int   i2 = s_redi[tid + s];
                if (v2 > s_red[tid] || (v2 == s_red[tid] && i2 < s_redi[tid])) {
                    s_red[tid] = v2; s_redi[tid] = i2;
                }
            }
            __syncthreads();
        }
        const float Mx  = s_red[0];
        const int   aidx = s_redi[0];

        if (tid < NNODES) atomicAdd(&s_sumexp, expf(s_scores[tid] - Mx));
        __syncthreads();

        if (tid == 0) {
            const int d = s_done;
            const int a = d ? 0 : aidx;
            if (!d) {
                s_logp += -logf(s_sumexp);   // logp[argmax] = -log(sum exp(s - max))
                if (a == 0) s_rem = s_capv;
                else { s_rem = s_rem - s_dem[a]; s_vis[a] = 1; }
                s_cur = a;
            }
            out_actions[step * BATCH + b] = d ? -1 : a;
        }
    }
    __syncthreads();
    if (tid == 0) out_logp[b] = s_logp;
}

extern "C" void kernel_launch(void* const* d_in, const int* in_sizes, int n_in,
                              void* d_out, int out_size, void* d_ws, size_t ws_size,
                              hipStream_t stream) {
    const float* coords = (const float*)d_in[0];
    const float* demands = (const float*)d_in[1];
    const float* capac  = (const float*)d_in[2];
    const float* We_w = (const float*)d_in[3];
    const float* We_b = (const float*)d_in[4];
    const float* Wq_w = (const float*)d_in[5];
    const float* Wq_b = (const float*)d_in[6];
    const float* Wk_w = (const float*)d_in[7];
    const float* Wk_b = (const float*)d_in[8];
    const float* Wv_w = (const float*)d_in[9];
    const float* Wv_b = (const float*)d_in[10];
    const float* P1_w = (const float*)d_in[11];
    const float* P1_b = (const float*)d_in[12];
    const float* P2_w = (const float*)d_in[13];
    // d_in[14] = P2_b (cancels in log_softmax), d_in[15] = max_steps (== 200)

    float* ws = (float*)d_ws;
    const size_t M = (size_t)BATCH * PB;
    float* gQ   = ws;
    float* gK   = ws + 1 * M;
    float* gV   = ws + 2 * M;
    float* gAtt = ws + 3 * M;
    float* gA1  = ws + 4 * M;

    ptrnet_encoder<<<BATCH, 256, 0, stream>>>(
        coords, demands, We_w, We_b, Wq_w, Wq_b, Wk_w, Wk_b, Wv_w, Wv_b,
        P1_w, P1_b, gQ, gK, gV, gAtt, gA1);

    int*   out_actions = (int*)d_out;
    float* out_logp    = (float*)d_out + STEPS * BATCH;
    ptrnet_decoder<<<BATCH, 256, 0, stream>>>(
        demands, capac, P1_w, P2_w, gAtt, gA1, out_actions, out_logp);
}